// RetinaHead_78443282694812
// MI455X (gfx1250) — compile-verified
//
#include <hip/hip_runtime.h>
#include <hip/hip_bf16.h>

#define TGT 64
#define NUM_CLASSES 80

// Shared helpers: MUST be the single source of truth for both passes so that
// pass-1 gt_max bits and pass-2 recomputed IoU bits are identical.
__device__ __forceinline__ void cvt_box(float cx, float cy, float w, float h, float g[4]) {
    g[0] = cx - 0.5f * w;
    g[1] = cy - 0.5f * h;
    g[2] = cx + 0.5f * w;
    g[3] = cy + 0.5f * h;
}

__device__ __forceinline__ float iou_fn(float ax1, float ay1, float ax2, float ay2, float aarea,
                                        float gx1, float gy1, float gx2, float gy2, float garea) {
    float lx = fmaxf(gx1, ax1);
    float ly = fmaxf(gy1, ay1);
    float rx = fminf(gx2, ax2);
    float ry = fminf(gy2, ay2);
    float w  = fmaxf(rx - lx, 0.0f);
    float h  = fmaxf(ry - ly, 0.0f);
    float inter = w * h;
    float uni = garea + aarea - inter;   // anchor area > 0 -> uni > 0, no NaN
    return inter / uni;
}

// ---------------------------------------------------------------------------
// Pass 1: one block per (image, gt). Scan all anchors (L2-resident, 480 KB),
// reduce max IoU with wave32 shuffles + LDS. Writes gt_max bit pattern.
// ---------------------------------------------------------------------------
__global__ __launch_bounds__(256) void gt_max_kernel(const float* __restrict__ anchors,
                                                     const float* __restrict__ tgt_boxes,
                                                     unsigned* __restrict__ gt_max_bits,
                                                     int A) {
    const int blk = blockIdx.x;                        // b*TGT + t
    const float4 raw = ((const float4*)tgt_boxes)[blk];
    float g[4];
    cvt_box(raw.x, raw.y, raw.z, raw.w, g);
    const float garea = (g[2] - g[0]) * (g[3] - g[1]);

    const float4* __restrict__ anc4 = (const float4*)anchors;
    float best = 0.0f;                                 // IoU >= 0 always
    for (int a = threadIdx.x; a < A; a += 256) {
        if (a + 4096 < A) __builtin_prefetch(&anc4[a + 4096], 0, 0);  // global_prefetch_b8
        float4 an = anc4[a];
        float aarea = (an.z - an.x) * (an.w - an.y);
        float v = iou_fn(an.x, an.y, an.z, an.w, aarea,
                         g[0], g[1], g[2], g[3], garea);
        best = fmaxf(best, v);
    }

    // wave32 reduction, then across the block's 8 waves
    #pragma unroll
    for (int m = 16; m > 0; m >>= 1) best = fmaxf(best, __shfl_xor(best, m, 32));
    __shared__ float wmax[8];
    if ((threadIdx.x & 31) == 0) wmax[threadIdx.x >> 5] = best;
    __syncthreads();
    if (threadIdx.x == 0) {
        float r = wmax[0];
        #pragma unroll
        for (int i = 1; i < 8; ++i) r = fmaxf(r, wmax[i]);
        gt_max_bits[blk] = __float_as_uint(r);
    }
}

// ---------------------------------------------------------------------------
// Pass 2: anchor-major. Stage raw GT boxes into LDS with the CDNA5 async
// global->LDS path (ASYNCcnt), convert once in LDS, then each thread matches
// one anchor against the 64 targets and emits label + deltas.
// ---------------------------------------------------------------------------
__global__ __launch_bounds__(256) void match_kernel(const float* __restrict__ anchors,
                                                    const float* __restrict__ tgt_boxes,
                                                    const int* __restrict__ tgt_labels,
                                                    const unsigned* __restrict__ gt_max_bits,
                                                    float* __restrict__ out_labels,
                                                    float* __restrict__ out_deltas,
                                                    int A) {
    __shared__ float    sraw[TGT * 4];    // raw cxcywh landed by async DMA
    __shared__ float    sbox[TGT][4];     // xyxy
    __shared__ float    sarea[TGT];
    __shared__ int      slab[TGT];
    __shared__ unsigned smax[TGT];        // per-gt max IoU bit pattern

    const int tid = threadIdx.x;
    const int b   = blockIdx.y;

    if (tid < TGT) {
        // LDS byte offset = low 32 bits of the flat shared-aperture address
        unsigned lds_addr = (unsigned)(size_t)&sraw[tid * 4];
        unsigned voff     = (unsigned)(tid * 16);
        const float* base = tgt_boxes + (size_t)b * TGT * 4;
        asm volatile("global_load_async_to_lds_b128 %0, %1, %2"
                     :: "v"(lds_addr), "v"(voff), "s"(base)
                     : "memory");
    }
    asm volatile("s_wait_asynccnt 0x0" ::: "memory");
    __syncthreads();

    if (tid < TGT) {
        float cx = sraw[tid * 4 + 0], cy = sraw[tid * 4 + 1];
        float w  = sraw[tid * 4 + 2], h  = sraw[tid * 4 + 3];
        float g[4];
        cvt_box(cx, cy, w, h, g);
        sbox[tid][0] = g[0]; sbox[tid][1] = g[1];
        sbox[tid][2] = g[2]; sbox[tid][3] = g[3];
        sarea[tid] = (g[2] - g[0]) * (g[3] - g[1]);
        slab[tid]  = tgt_labels[b * TGT + tid];
        smax[tid]  = gt_max_bits[b * TGT + tid];
    }
    __syncthreads();

    const int a = blockIdx.x * 256 + tid;
    if (a >= A) return;

    const float4 an = ((const float4*)anchors)[a];
    const float aarea = (an.z - an.x) * (an.w - an.y);

    float best = -1.0f;      // strict '>' keeps first occurrence == jnp.argmax
    int   bid  = 0;
    bool  high = false;
    #pragma unroll 8
    for (int t = 0; t < TGT; ++t) {
        float v = iou_fn(an.x, an.y, an.z, an.w, aarea,
                         sbox[t][0], sbox[t][1], sbox[t][2], sbox[t][3], sarea[t]);
        high = high || (__float_as_uint(v) == smax[t]);   // exact: same bits both passes
        if (v > best) { best = v; bid = t; }
    }

    // Matcher(0.4, 0.5, allow_low_quality_matches)
    int match = high ? 1 : (best >= 0.5f ? 1 : (best >= 0.4f ? -1 : 0));
    int lab   = (match == 1) ? slab[bid] : ((match == 0) ? NUM_CLASSES : -1);
    out_labels[(size_t)b * A + a] = (float)lab;

    // Box2BoxTransform.get_deltas, weights (1,1,1,1)
    float sw = an.z - an.x, shh = an.w - an.y;
    float sx = an.x + 0.5f * sw, sy = an.y + 0.5f * shh;
    float gx1 = sbox[bid][0], gy1 = sbox[bid][1];
    float gx2 = sbox[bid][2], gy2 = sbox[bid][3];
    float tw = gx2 - gx1, th = gy2 - gy1;
    float tx = gx1 + 0.5f * tw, ty = gy1 + 0.5f * th;
    float4 d;
    d.x = (tx - sx) / sw;
    d.y = (ty - sy) / shh;
    d.z = logf(tw / sw);
    d.w = logf(th / shh);
    ((float4*)out_deltas)[(size_t)b * A + a] = d;
}

extern "C" void kernel_launch(void* const* d_in, const int* in_sizes, int n_in,
                              void* d_out, int out_size, void* d_ws, size_t ws_size,
                              hipStream_t stream) {
    const float* anchors    = (const float*)d_in[0];   // [A,4]
    const float* tgt_boxes  = (const float*)d_in[1];   // [B,T,4] cxcywh
    const int*   tgt_labels = (const int*)d_in[2];     // [B,T]

    const int A    = in_sizes[0] / 4;
    const int BT   = in_sizes[2];          // B*T
    const int Bimg = BT / TGT;

    unsigned* gt_max_bits = (unsigned*)d_ws;           // B*T uints
    float* out        = (float*)d_out;
    float* out_labels = out;                           // [B,A]
    float* out_deltas = out + (size_t)Bimg * A;        // [B,A,4]

    gt_max_kernel<<<BT, 256, 0, stream>>>(anchors, tgt_boxes, gt_max_bits, A);

    dim3 grid((A + 255) / 256, Bimg);
    match_kernel<<<grid, 256, 0, stream>>>(anchors, tgt_boxes, tgt_labels, gt_max_bits,
                                           out_labels, out_deltas, A);
}